// I1Pool_326417514933
// MI455X (gfx1250) — compile-verified
//
#include <hip/hip_runtime.h>

// Gated directional max-pool scan along W (LBYLNet landmark pooling).
//   h_0 = x_0 ; h_i = max(x_i, g * h_{i-1})  along W (innermost, stride 1).
// Memory-bound: 268 MB total traffic -> ~11.5us floor at 23.3 TB/s.
//
// One wave32 per row of W=64 floats: lane l owns elements {2l, 2l+1}.
// The recurrence composes as f(h)=max(a, b*h):
//   (a2,b2) o (a1,b1) = (max(a2, b2*a1), b2*b1)   (associative)
// -> 5-step Hillis-Steele inclusive scan across lanes with shuffles.
// After the scan, lane l's A equals h_{2l+1}; h_{2l} uses lane (l-1)'s prefix.
//
// Grid encodes the guide index: blockIdx.y = (b*C + c), so guide[] is read
// with a block-uniform index (scalar load) and no integer division appears.

#define WAVES_PER_BLOCK 8
#define THREADS_PER_BLOCK (WAVES_PER_BLOCK * 32)

__global__ __launch_bounds__(THREADS_PER_BLOCK)
void i1pool_wave_scan(const float* __restrict__ x,
                      const float* __restrict__ guide,
                      float* __restrict__ out,
                      int H)
{
    const int waveInBlk   = threadIdx.x >> 5;
    const int lane        = threadIdx.x & 31;
    const int waveInGuide = blockIdx.x * WAVES_PER_BLOCK + waveInBlk;
    if (waveInGuide >= H) return;                   // wave-uniform guard

    // Block-uniform guide index -> scalar load.
    const float g = guide[blockIdx.y];

    const size_t row = (size_t)blockIdx.y * H + waveInGuide;

    // Coalesced b64 load: the wave covers exactly the 256B row.
    const float2* __restrict__ xrow =
        reinterpret_cast<const float2*>(x + row * 64);
    const float2 v = xrow[lane];

    // Per-lane pair composed into one segment function (A, B):
    //   A = max(x1, g*x0)   (value term of the 2-element segment)
    //   B = g*g             (carry-through factor of the segment)
    float A = fmaxf(v.y, g * v.x);
    float B = g * g;

    // Inclusive scan over lanes of the (max, *) composition.
    // Predicated update (no -inf identity needed -> no 0*inf NaN when g==0).
#pragma unroll
    for (int d = 1; d < 16; d <<= 1) {
        const float Au = __shfl_up(A, d, 32);
        const float Bu = __shfl_up(B, d, 32);
        if (lane >= d) {
            A = fmaxf(A, B * Au);
            B = B * Bu;
        }
    }
    {   // Last step: B is never consumed afterwards -> skip its shuffle.
        const float Au = __shfl_up(A, 16, 32);
        if (lane >= 16) A = fmaxf(A, B * Au);
    }

    // A now equals h_{2*lane+1}. Exclusive prefix gives h_{2*lane}.
    const float Ap = __shfl_up(A, 1, 32);           // h_{2*lane-1} for lane>0
    float h0 = v.x;
    if (lane > 0) h0 = fmaxf(v.x, g * Ap);

    float2 o;
    o.x = h0;
    o.y = A;
    reinterpret_cast<float2*>(out + row * 64)[lane] = o;
}

extern "C" void kernel_launch(void* const* d_in, const int* in_sizes, int n_in,
                              void* d_out, int out_size, void* d_ws, size_t ws_size,
                              hipStream_t stream) {
    (void)n_in; (void)out_size; (void)d_ws; (void)ws_size;

    const float* x     = (const float*)d_in[0];   // [B,C,H,W] f32
    const float* guide = (const float*)d_in[1];   // [B,C,1,1] f32
    float* out         = (float*)d_out;

    const int W          = 64;                    // reference innermost dim
    const int rows       = in_sizes[0] / W;       // B*C*H = 524288
    const int guideRows  = in_sizes[1];           // B*C   = 8192
    const int H          = rows / guideRows;      // 64

    const int blocksX = (H + WAVES_PER_BLOCK - 1) / WAVES_PER_BLOCK;  // 8
    dim3 grid(blocksX, guideRows);                                    // (8, 8192)
    i1pool_wave_scan<<<grid, THREADS_PER_BLOCK, 0, stream>>>(x, guide, out, H);
}